// bicon_loss_61684320305394
// MI455X (gfx1250) — compile-verified
//
#include <hip/hip_runtime.h>
#include <stdint.h>

// Problem constants (match reference)
#define BATCH 16
#define HH 352
#define WW 352
#define CH 8
#define TH 8                       // interior rows per tile
#define TR (TH + 2)                // rows incl. +-1 halo
#define TILE_F (CH * TR * WW)      // 28160 floats in LDS (112,640 B)
#define NPIX (TH * WW)             // 2816 interior pixels per tile (= 11*256)
#define NTILES (HH / TH)           // 44
#define NBLOCKS (BATCH * NTILES)   // 704

#define LN2f 0.6931471805599453f
#define L2Ef 1.4426950408889634f
#define EPSf 1e-7f

typedef __attribute__((ext_vector_type(4))) unsigned u32x4;
typedef __attribute__((ext_vector_type(8))) unsigned u32x8;

__device__ __forceinline__ float fsigmoid(float x) {
    // 1/(1+e^-x) via native v_exp_f32 (2^x) + v_rcp_f32
    float e = __builtin_amdgcn_exp2f(-x * L2Ef);
    return __builtin_amdgcn_rcpf(1.0f + e);
}
__device__ __forceinline__ float flog(float x) {   // natural log via v_log_f32
    return __builtin_amdgcn_logf(x) * LN2f;
}
__device__ __forceinline__ float fclip(float p) {
    return fminf(fmaxf(p, EPSf), 1.0f - EPSf);
}
// -[t*ln(p) + (1-t)*ln(1-p)] for binary t: one log via operand select
__device__ __forceinline__ float bce_bin(float p, bool t1) {
    float pc = fclip(p);
    float arg = t1 ? pc : (1.0f - pc);
    return -flog(arg);
}

__global__ __launch_bounds__(256)
void bicon_tile_kernel(const float* __restrict__ cmap,
                       const float* __restrict__ targ,
                       const int*   __restrict__ conin,
                       float* __restrict__ partials)
{
    __shared__ float tile[TILE_F];   // 8 ch x 10 rows x 352 cols (p after phase 2)
    __shared__ float wsum[8];

    const int tid = threadIdx.x;
    const int blk = blockIdx.x;
    const int b   = blk / NTILES;
    const int h0  = (blk % NTILES) * TH;

    // ---- Phase 1: TDM tensor_load_to_lds, one 2D tile per channel, one per wave ----
    // Each of the 8 waves DMAs its channel's (up to) 10x352 f32 sub-tensor into LDS.
    {
        const unsigned lds_base = (unsigned)(uintptr_t)tile; // low 32 bits = LDS offset
        const int wv    = tid >> 5;                 // wave id == channel id (8 waves)
        const int first = (h0 - 1 < 0) ? 0 : (h0 - 1);
        const int last  = (h0 + TH > HH - 1) ? (HH - 1) : (h0 + TH);
        const unsigned nrows    = (unsigned)(last - first + 1);   // 9 or 10
        const unsigned lds_row0 = (unsigned)(first - (h0 - 1));   // 0 or 1
        const unsigned lds_addr =
            lds_base + (unsigned)(((wv * TR + (int)lds_row0) * WW) * 4);
        const uint64_t ga = (uint64_t)(uintptr_t)cmap
            + (uint64_t)((((b * CH + wv) * HH + first) * WW)) * 4ull;

        // D# group 0: count=1, lds_addr, 57-bit global addr, type=2 ("image")
        u32x4 g0;
        g0[0] = 1u;
        g0[1] = lds_addr;
        g0[2] = (unsigned)ga;
        g0[3] = ((unsigned)(ga >> 32) & 0x01FFFFFFu) | (2u << 30);

        // D# group 1: data_size=4B; 2D tile: dim0=352 cols, dim1=nrows; stride0=352
        u32x8 g1;
        g1[0] = 0x00020000u;                 // data_size=2 (4 bytes), all flags 0
        g1[1] = ((unsigned)WW) << 16;        // tensor_dim0[15:0] = 352
        g1[2] = (nrows & 0xFFFFu) << 16;     // tensor_dim0[31:16]=0 | tensor_dim1[15:0]
        g1[3] = ((unsigned)WW) << 16;        // tensor_dim1[31:16]=0 | tile_dim0 = 352
        g1[4] = nrows;                       // tile_dim1 = nrows, tile_dim2 = 0 (2D)
        g1[5] = (unsigned)WW;                // tensor_dim0_stride[31:0] = 352
        g1[6] = 0u;                          // stride0 hi | stride1 lo (unused, 2D)
        g1[7] = 0u;

        asm volatile("tensor_load_to_lds %0, %1" :: "s"(g0), "s"(g1) : "memory");
        __builtin_amdgcn_s_wait_tensorcnt(0);
    }
    __syncthreads();

    // ---- Phase 2: in-place sigmoid over the tile; zero off-image halo rows ----
    #pragma unroll 4
    for (int k = 0; k < TILE_F / 256; ++k) {    // 110 iterations, exact
        int idx = tid + k * 256;
        tile[idx] = fsigmoid(tile[idx]);
    }
    {
        const bool top = (h0 == 0);
        const bool bot = (h0 + TH == HH);
        if (top | bot) {                        // boundary bands only (2 of 44)
            __syncthreads();
            for (int k = 0; k < (CH * WW) / 256; ++k) {   // 11 iterations
                int idx = tid + k * 256;
                int c = idx / WW;
                int w = idx - c * WW;
                if (top) tile[(c * TR + 0) * WW + w]        = 0.0f;
                if (bot) tile[(c * TR + (TR - 1)) * WW + w] = 0.0f;
            }
        }
    }
    __syncthreads();

    // ---- Phase 3: fused vote + three BCE sums over interior pixels ----
    // shifted[c](h,w) = p[7-c](h - dy_c, w - dx_c), zero padded
    const int DX[8] = { 1, 0, -1, 1, -1, 1, 0, -1 };
    const int DY[8] = { 1, 1,  1, 0,  0,-1,-1, -1 };

    float acc = 0.0f;
    for (int k = 0; k < NPIX / 256; ++k) {      // 11 iterations, exact
        int q  = tid + k * 256;
        int w  = q % WW;
        int rr = q / WW;                        // 0..7
        int r  = rr + 1;                        // LDS row
        int h  = h0 + rr;                       // global row

        float sumcon = 0.0f;
        float vmax = -3.0e38f, vmin = 3.0e38f;
        #pragma unroll
        for (int c = 0; c < CH; ++c) {
            float p = tile[(c * TR + r) * WW + w];
            int wc  = w - DX[c];
            float s = ((unsigned)wc < (unsigned)WW)
                        ? tile[((7 - c) * TR + (r - DY[c])) * WW + wc] : 0.0f;
            float v = p * s;
            int cv  = __builtin_nontemporal_load(&conin[((b * CH + c) * HH + h) * WW + w]);
            bool t1 = (cv != 0);
            sumcon += t1 ? 1.0f : 0.0f;
            vmax = fmaxf(vmax, v);
            vmin = fminf(vmin, v);
            acc += 0.8f * bce_bin(p, t1) + 0.2f * bce_bin(v, t1);
        }
        bool edge = (sumcon > 0.5f) && (sumcon < 7.5f);
        float dec = edge ? (1.0f - vmin) : vmax;
        float dc  = fclip(dec);
        float tt  = __builtin_nontemporal_load(&targ[(b * HH + h) * WW + w]);
        acc += -(tt * flog(dc) + (1.0f - tt) * flog(1.0f - dc));
    }

    // ---- Block reduction (wave32 shuffles -> LDS -> one partial per block) ----
    #pragma unroll
    for (int off = 16; off > 0; off >>= 1)
        acc += __shfl_xor(acc, off, 32);
    if ((tid & 31) == 0) wsum[tid >> 5] = acc;
    __syncthreads();
    if (tid < 8) {
        float v = wsum[tid];
        v += __shfl_xor(v, 4, 32);
        v += __shfl_xor(v, 2, 32);
        v += __shfl_xor(v, 1, 32);
        if (tid == 0) partials[blk] = v;
    }
}

__global__ __launch_bounds__(256)
void bicon_reduce_kernel(const float* __restrict__ partials, int n,
                         float* __restrict__ out)
{
    __shared__ float wsum[8];
    float a = 0.0f;
    for (int i = threadIdx.x; i < n; i += 256) a += partials[i];
    #pragma unroll
    for (int off = 16; off > 0; off >>= 1)
        a += __shfl_xor(a, off, 32);
    if ((threadIdx.x & 31) == 0) wsum[threadIdx.x >> 5] = a;
    __syncthreads();
    if (threadIdx.x < 8) {
        float v = wsum[threadIdx.x];
        v += __shfl_xor(v, 4, 32);
        v += __shfl_xor(v, 2, 32);
        v += __shfl_xor(v, 1, 32);
        if (threadIdx.x == 0) out[0] = v;
    }
}

extern "C" void kernel_launch(void* const* d_in, const int* in_sizes, int n_in,
                              void* d_out, int out_size, void* d_ws, size_t ws_size,
                              hipStream_t stream) {
    (void)in_sizes; (void)n_in; (void)out_size; (void)ws_size;
    const float* cmap  = (const float*)d_in[0];
    const float* targ  = (const float*)d_in[1];
    const int*   conin = (const int*)d_in[2];
    float* out      = (float*)d_out;
    float* partials = (float*)d_ws;   // NBLOCKS floats of scratch

    bicon_tile_kernel<<<NBLOCKS, 256, 0, stream>>>(cmap, targ, conin, partials);
    bicon_reduce_kernel<<<1, 256, 0, stream>>>(partials, NBLOCKS, out);
}